// BiLSTM_CRF_60138132079323
// MI455X (gfx1250) — compile-verified
//
#include <hip/hip_runtime.h>
#include <hip/hip_bf16.h>
#include <math.h>
#include <stdint.h>

// ---------------------------------------------------------------------------
// BiLSTM-CRF on MI455X (gfx1250), wave32.
//  - f32 WMMA (V_WMMA_F32_16X16X4_F32) for all matrix ops (exact f32 math).
//  - Persistent per-direction LSTM workgroup: h in LDS, c in registers,
//    per-step gate slice (128 KB, time-major) prefetched by the Tensor Data
//    Mover into a double-buffered LDS staging area (TENSORcnt overlap).
// Sizes: B=32, S=256, E=256, H=256/dir, 4H=1024 gates, TAGS=12.
// ---------------------------------------------------------------------------

typedef float v2f __attribute__((ext_vector_type(2)));
typedef float v8f __attribute__((ext_vector_type(8)));
typedef unsigned int u32;
typedef unsigned int v4u __attribute__((ext_vector_type(4)));
typedef int v4i __attribute__((ext_vector_type(4)));
typedef int v8i __attribute__((ext_vector_type(8)));

#define WMMA_F32(a, b, c) \
  __builtin_amdgcn_wmma_f32_16x16x4_f32(false, (a), false, (b), (short)0, (c), false, false)

#if defined(__has_builtin)
#if __has_builtin(__builtin_amdgcn_tensor_load_to_lds)
#define HAVE_TDM 1
#endif
#endif
#ifndef HAVE_TDM
#define HAVE_TDM 0
#endif
#if __has_include(<hip/amd_detail/amd_gfx1250_TDM.h>)
#define TDM_6ARG 1
#else
#define TDM_6ARG 0
#endif

__device__ __forceinline__ float sigmoidf_(float x) {
  return 1.0f / (1.0f + expf(-x));
}

#if HAVE_TDM
// DMA a contiguous 131072-byte (32768 x f32) block from global to LDS.
// D# group0: count=1; lds_addr; 57-bit global addr; type=2 ("image").
// D# group1: data_size=4B, tensor_dim0=32768, tensor_dim1=1, tile_dim0=32768,
//            tensor_dim0_stride=32768, no padding / multicast / iterate.
__device__ __forceinline__ void tdm_load_slice(const float* gptr, u32 lds_off) {
  unsigned long long ga = (unsigned long long)(uintptr_t)gptr;
  v4u g0;
  g0.x = 1u;                                              // count=1
  g0.y = lds_off;                                         // lds_addr
  g0.z = (u32)ga;                                         // global_addr[31:0]
  g0.w = ((u32)(ga >> 32) & 0x01FFFFFFu) | 0x80000000u;   // addr[56:32]|type=2
  v8i g1;
  g1[0] = 0x00020000;        // workgroup_mask=0, data_size=2 (4 bytes)
  g1[1] = (int)0x80000000;   // tensor_dim0[15:0]=32768 in bits [63:48]
  g1[2] = 0x00010000;        // tensor_dim0[31:16]=0, tensor_dim1=1
  g1[3] = (int)0x80000000;   // tile_dim0=32768 in bits [127:112]
  g1[4] = 0;                 // tile_dim1=0 (unused), tile_dim2=0
  g1[5] = 32768;             // tensor_dim0_stride[31:0]
  g1[6] = 0;
  g1[7] = 0;
  v4i z4 = {0, 0, 0, 0};
#if TDM_6ARG
  v8i z8 = {0, 0, 0, 0, 0, 0, 0, 0};
  __builtin_amdgcn_tensor_load_to_lds(g0, g1, z4, z4, z8, 0);
#else
  __builtin_amdgcn_tensor_load_to_lds(g0, g1, z4, z4, 0);
#endif
}
#endif  // HAVE_TDM

// ---------------------------------------------------------------------------
// Kernel 1: embedding gather  x[row][k] = emb[sentences[row]][k]
// ---------------------------------------------------------------------------
__global__ __launch_bounds__(256) void k_gather(const int* __restrict__ sent,
                                                const float* __restrict__ emb,
                                                float* __restrict__ x) {
  int row = blockIdx.x;            // 0..8191 (b*S + s)
  int k = threadIdx.x;             // 0..255
  int tok = sent[row];
  x[(size_t)row * 256 + k] = emb[(size_t)tok * 256 + k];
}

// ---------------------------------------------------------------------------
// Kernel 2: xg GEMM, output TIME-MAJOR: xg[dir][time][b][g]
//   xg = x . wih^T + bih + bhh.  One wave per 16x16 tile, K=256 -> 64 WMMA.
//   waves = 2 dirs * 512 mtiles * 64 ntiles = 65536 -> 8192 blocks of 256.
// ---------------------------------------------------------------------------
__global__ __launch_bounds__(256) void k_xg_gemm(
    const float* __restrict__ x,
    const float* __restrict__ wih_f, const float* __restrict__ bih_f,
    const float* __restrict__ bhh_f,
    const float* __restrict__ wih_b, const float* __restrict__ bih_b,
    const float* __restrict__ bhh_b,
    float* __restrict__ xg) {
  int wave = (blockIdx.x * blockDim.x + threadIdx.x) >> 5;
  int lane = threadIdx.x & 31;
  int ntile = wave & 63;
  int mtile = (wave >> 6) & 511;
  int dir = wave >> 15;

  const float* wih = dir ? wih_b : wih_f;
  const float* bih = dir ? bih_b : bih_f;
  const float* bhh = dir ? bhh_b : bhh_f;

  int m0 = mtile * 16, n0 = ntile * 16;
  int hi = lane >> 4;
  int l = lane & 15;

  const float* arow = x + (size_t)(m0 + l) * 256 + (hi ? 2 : 0);
  const float* brow = wih + (size_t)(n0 + l) * 256 + (hi ? 2 : 0);

  v8f acc = {};
#pragma unroll 8
  for (int k0 = 0; k0 < 256; k0 += 4) {
    v2f a = *(const v2f*)(arow + k0);
    v2f b = *(const v2f*)(brow + k0);
    acc = WMMA_F32(a, b, acc);
  }

  float bias = bih[n0 + l] + bhh[n0 + l];
  float* out = xg + (size_t)dir * 8192 * 1024;
#pragma unroll
  for (int r = 0; r < 8; ++r) {
    int m = m0 + r + hi * 8;       // row index b*256 + s
    int bb = m >> 8, tt = m & 255;
    out[((size_t)tt * 32 + bb) * 1024 + n0 + l] = acc[r] + bias;  // time-major
  }
}

// ---------------------------------------------------------------------------
// Kernel 3: persistent LSTM recurrence. grid = 2 (dir), block = 1024 (32 waves).
// LDS: h[32][256] (32 KB) + sgx[2][32][1024] (256 KB) = 288 KB of 320 KB.
// Cell state c lives in 8 VGPRs/thread (fixed (b,j) ownership across steps).
// Per step t:
//   wave0: s_wait_tensorcnt(0)  -> slice t resident in sgx[t&1]
//   barrier
//   wave0: TDM-issue slice t+1 -> sgx[~t&1]   (overlaps the whole step)
//   all:   G-tile WMMA (h @ whh^T) from LDS h + L2-resident whh,
//          accumulate into sgx[t&1] (already holds xg + biases)
//   barrier
//   all:   gate nonlinearities, c update (regs), h -> LDS + global
// ---------------------------------------------------------------------------
__global__ __launch_bounds__(1024) void k_lstm(const float* __restrict__ xg,
                                               const float* __restrict__ whh_f,
                                               const float* __restrict__ whh_b,
                                               float* __restrict__ hs) {
  __shared__ float sh[32 * 256];        // h (32 KB)
  __shared__ float sgx[2][32 * 1024];   // staged xg slice / gate accum (256 KB)

  int dir = blockIdx.x;
  const float* whh = dir ? whh_b : whh_f;
  const float* xgd = xg + (size_t)dir * 8192 * 1024;
  float* hsd = hs + (size_t)dir * 8192 * 256;

  int tid = threadIdx.x;
  for (int i = tid; i < 32 * 256; i += 1024) sh[i] = 0.0f;

  int wave = tid >> 5, lane = tid & 31;
  int hi = lane >> 4, l = lane & 15;
  int n0w = wave * 32;

  float creg[8];
#pragma unroll
  for (int p = 0; p < 8; ++p) creg[p] = 0.0f;

#if HAVE_TDM
  if (wave == 0) {
    int time0 = dir ? 255 : 0;
    tdm_load_slice(xgd + (size_t)time0 * 32768, (u32)(uintptr_t)(&sgx[0][0]));
  }
#endif

  for (int t = 0; t < 256; ++t) {
    int time = dir ? (255 - t) : t;
#if HAVE_TDM
    int cur = t & 1;
    if (wave == 0) __builtin_amdgcn_s_wait_tensorcnt(0);
    __syncthreads();  // slice t resident; h(t-1) visible; prev reads done
    if (wave == 0 && t < 255) {
      int timen = dir ? (254 - t) : (t + 1);
      tdm_load_slice(xgd + (size_t)timen * 32768,
                     (u32)(uintptr_t)(&sgx[cur ^ 1][0]));
    }
#else
    int cur = 0;
    __syncthreads();  // prev step's sgx reads done; h(t-1) visible
    for (int i = tid; i < 32 * 1024; i += 1024)
      sgx[0][i] = xgd[(size_t)time * 32768 + i];
    __syncthreads();
#endif

    // --- G tiles: two 16-col tiles per wave, batches 0-15 / 16-31 ---
#pragma unroll
    for (int nt = 0; nt < 2; ++nt) {
      int n0 = n0w + nt * 16;
      v8f acc0 = {}, acc1 = {};
      const float* brow = whh + (size_t)(n0 + l) * 256 + (hi ? 2 : 0);
      const float* a0 = sh + (0 + l) * 256 + (hi ? 2 : 0);
      const float* a1 = sh + (16 + l) * 256 + (hi ? 2 : 0);
#pragma unroll 4
      for (int k0 = 0; k0 < 256; k0 += 4) {
        v2f b = *(const v2f*)(brow + k0);
        v2f av0 = *(const v2f*)(a0 + k0);
        v2f av1 = *(const v2f*)(a1 + k0);
        acc0 = WMMA_F32(av0, b, acc0);
        acc1 = WMMA_F32(av1, b, acc1);
      }
#pragma unroll
      for (int r = 0; r < 8; ++r) {
        int mA = r + hi * 8;         // batches 0..15
        int mB = 16 + r + hi * 8;    // batches 16..31
        sgx[cur][mA * 1024 + n0 + l] += acc0[r];
        sgx[cur][mB * 1024 + n0 + l] += acc1[r];
      }
    }
    __syncthreads();  // gates complete

    // --- elementwise gate update: 8 fixed (b,j) pairs per thread ---
#pragma unroll
    for (int p = 0; p < 8; ++p) {
      int idx = tid + p * 1024;
      int b = idx >> 8;
      int j = idx & 255;
      const float* gb = &sgx[cur][b * 1024];
      float gi = gb[j];
      float gf = gb[256 + j];
      float gg = gb[512 + j];
      float go = gb[768 + j];
      float c = sigmoidf_(gf) * creg[p] + sigmoidf_(gi) * tanhf(gg);
      float h = sigmoidf_(go) * tanhf(c);
      creg[p] = c;
      sh[b * 256 + j] = h;
      hsd[((size_t)b * 256 + time) * 256 + j] = h;
    }
  }
}

// ---------------------------------------------------------------------------
// Kernel 4: feats[row][t] = [hf|hb][row][:] . w_out[t][:] + b_out[t], t<12.
// One wave per 16-row tile, N padded to 16 (cols 12..15 multiply zero).
// ---------------------------------------------------------------------------
__global__ __launch_bounds__(256) void k_proj(const float* __restrict__ hs,
                                              const float* __restrict__ w_out,
                                              const float* __restrict__ b_out,
                                              float* __restrict__ feats) {
  int wave = (blockIdx.x * blockDim.x + threadIdx.x) >> 5;  // 0..511
  int lane = threadIdx.x & 31;
  int hi = lane >> 4, l = lane & 15;
  int m0 = wave * 16;

  const float* h0 = hs + (size_t)(m0 + l) * 256 + (hi ? 2 : 0);          // fwd
  const float* h1 = hs + (size_t)(8192 + m0 + l) * 256 + (hi ? 2 : 0);   // bwd
  bool valid = (l < 12);
  const float* wrow = w_out + (size_t)(valid ? l : 0) * 512 + (hi ? 2 : 0);
  v2f zero = {0.0f, 0.0f};

  v8f acc = {};
#pragma unroll 4
  for (int k0 = 0; k0 < 256; k0 += 4) {       // forward half (w_out[:, :256])
    v2f a = *(const v2f*)(h0 + k0);
    v2f b = valid ? *(const v2f*)(wrow + k0) : zero;
    acc = WMMA_F32(a, b, acc);
  }
#pragma unroll 4
  for (int k0 = 0; k0 < 256; k0 += 4) {       // backward half (w_out[:, 256:])
    v2f a = *(const v2f*)(h1 + k0);
    v2f b = valid ? *(const v2f*)(wrow + 256 + k0) : zero;
    acc = WMMA_F32(a, b, acc);
  }

  if (valid) {
    float bb = b_out[l];
#pragma unroll
    for (int r = 0; r < 8; ++r) {
      int m = m0 + r + hi * 8;
      feats[(size_t)m * 12 + l] = acc[r] + bb;
    }
  }
}

// ---------------------------------------------------------------------------
// Kernel 5: Viterbi. One wave per sequence (32 blocks x 32 threads).
// out layout: [0..31] scores (f32), [32..32+8191] path tags (as f32 values).
// ---------------------------------------------------------------------------
__global__ __launch_bounds__(32) void k_viterbi(const float* __restrict__ feats,
                                                const float* __restrict__ trans,
                                                float* __restrict__ out) {
  __shared__ float fv[16];
  __shared__ float tr[12][12];
  __shared__ int bp[256][12];

  int b = blockIdx.x, lane = threadIdx.x;
  for (int i = lane; i < 144; i += 32) tr[i / 12][i % 12] = trans[i];
  if (lane < 12) fv[lane] = (lane == 10) ? 0.0f : -10000.0f;  // START_IDX=10
  __syncthreads();

  for (int s = 0; s < 256; ++s) {
    float nf = 0.0f;
    int bestp = 0;
    if (lane < 12) {
      float best = -3.402823e38f;
      for (int p = 0; p < 12; ++p) {
        float v = fv[p] + tr[lane][p];
        if (v > best) { best = v; bestp = p; }
      }
      nf = best + feats[((size_t)b * 256 + s) * 12 + lane];
      bp[s][lane] = bestp;
    }
    __syncthreads();
    if (lane < 12) fv[lane] = nf;
    __syncthreads();
  }

  if (lane == 0) {
    float best = -3.402823e38f;
    int bt = 0;
    for (int t2 = 0; t2 < 12; ++t2) {
      float v = fv[t2] + tr[11][t2];  // STOP_IDX=11
      if (v > best) { best = v; bt = t2; }
    }
    out[b] = best;
    float* path = out + 32 + (size_t)b * 256;
    int tag = bt;
    path[255] = (float)tag;
    for (int s = 254; s >= 0; --s) {
      tag = bp[s + 1][tag];
      path[s] = (float)tag;
    }
  }
}

// ---------------------------------------------------------------------------
// Launch
// ---------------------------------------------------------------------------
extern "C" void kernel_launch(void* const* d_in, const int* in_sizes, int n_in,
                              void* d_out, int out_size, void* d_ws,
                              size_t ws_size, hipStream_t stream) {
  (void)in_sizes; (void)n_in; (void)out_size; (void)ws_size;

  const int* sentences = (const int*)d_in[0];
  const float* emb   = (const float*)d_in[1];
  const float* wih_f = (const float*)d_in[2];
  const float* whh_f = (const float*)d_in[3];
  const float* bih_f = (const float*)d_in[4];
  const float* bhh_f = (const float*)d_in[5];
  const float* wih_b = (const float*)d_in[6];
  const float* whh_b = (const float*)d_in[7];
  const float* bih_b = (const float*)d_in[8];
  const float* bhh_b = (const float*)d_in[9];
  const float* w_out = (const float*)d_in[10];
  const float* b_out = (const float*)d_in[11];
  const float* trans = (const float*)d_in[12];

  char* ws = (char*)d_ws;
  float* X  = (float*)(ws);                       //  8 MB  [8192][256]
  float* XG = (float*)(ws + (8ull  << 20));       // 64 MB  [2][256][32][1024]
  float* HS = (float*)(ws + (72ull << 20));       // 16 MB  [2][8192][256]
  float* FE = (float*)(ws + (88ull << 20));       // 384 KB [8192][12]

  k_gather<<<8192, 256, 0, stream>>>(sentences, emb, X);
  k_xg_gemm<<<8192, 256, 0, stream>>>(X, wih_f, bih_f, bhh_f,
                                      wih_b, bih_b, bhh_b, XG);
  k_lstm<<<2, 1024, 0, stream>>>(XG, whh_f, whh_b, HS);
  k_proj<<<64, 256, 0, stream>>>(HS, w_out, b_out, FE);
  k_viterbi<<<32, 32, 0, stream>>>(FE, trans, (float*)d_out);
}